// SinkhornDistance_72172630442644
// MI455X (gfx1250) — compile-verified
//
#include <hip/hip_runtime.h>
#include <stdint.h>

// Sinkhorn with C = Identity(64): the [B,64,64] kernel tensor collapses to
// O(N)-per-row updates. One wave32 per batch row, 2 elements per lane.

#define N_DIM      64
#define WAVES_PER_BLOCK 8
#define EPS_F      0.1f
#define INV_EPS    10.0f
#define MAX_ITER   100
#define TOL_F      1e-3f
#define LOG1EM8   -18.420680743952367f   // log(1e-8)
#define ONE_M_EM10 0.9999546000702375f   // 1 - exp(-1/eps) = 1 - exp(-10)

__device__ __forceinline__ float wave_max32(float v) {
#pragma unroll
  for (int m = 16; m >= 1; m >>= 1) v = fmaxf(v, __shfl_xor(v, m, 32));
  return v;
}
__device__ __forceinline__ float wave_sum32(float v) {
#pragma unroll
  for (int m = 16; m >= 1; m >>= 1) v += __shfl_xor(v, m, 32);
  return v;
}
// Wave-uniform broadcast of lane k: v_readlane_b32 (VALU->SGPR), cheaper than
// ds_bpermute for a uniform index.
__device__ __forceinline__ float bcast_lane(float v, int k) {
  return __int_as_float(__builtin_amdgcn_readlane(__float_as_int(v), k));
}

__global__ __launch_bounds__(256) void sinkhorn_rows_kernel(
    const float* __restrict__ x, const float* __restrict__ y,
    float* __restrict__ block_partial, int B) {
  __shared__ float s_x[WAVES_PER_BLOCK][N_DIM];
  __shared__ float s_y[WAVES_PER_BLOCK][N_DIM];
  __shared__ float s_w[WAVES_PER_BLOCK];

  const int lane = threadIdx.x & 31;
  const int wv   = threadIdx.x >> 5;
  const int row  = blockIdx.x * WAVES_PER_BLOCK + wv;

  float wrow = 0.0f;
  if (row < B) {
    // ---- Stage this wave's x/y row into LDS via the gfx1250 async copy path.
    // Each lane moves 8 bytes (b64); ASYNCcnt-tracked, then s_wait_asynccnt 0.
    const float* gx = x + (size_t)row * N_DIM + lane * 2;
    const float* gy = y + (size_t)row * N_DIM + lane * 2;
    uint32_t lx = (uint32_t)(uintptr_t)&s_x[wv][lane * 2];
    uint32_t ly = (uint32_t)(uintptr_t)&s_y[wv][lane * 2];
    asm volatile(
        "global_load_async_to_lds_b64 %0, %2, off\n\t"
        "global_load_async_to_lds_b64 %1, %3, off\n\t"
        "s_wait_asynccnt 0x0"
        :: "v"(lx), "v"(ly), "v"(gx), "v"(gy)
        : "memory");

    // Lane holds elements (lane) and (lane+32) of the 64-wide row.
    float x0 = s_x[wv][lane], x1 = s_x[wv][lane + 32];
    float y0 = s_y[wv][lane], y1 = s_y[wv][lane + 32];

    // ---- log(clip(softmax(x), 1e-8, 1)) = max(x - m - log(sum), log 1e-8)
    float mx  = wave_max32(fmaxf(x0, x1));
    float lsx = __logf(wave_sum32(__expf(x0 - mx) + __expf(x1 - mx)));
    float logx0 = fmaxf(x0 - mx - lsx, LOG1EM8);
    float logx1 = fmaxf(x1 - mx - lsx, LOG1EM8);

    float my  = wave_max32(fmaxf(y0, y1));
    float lsy = __logf(wave_sum32(__expf(y0 - my) + __expf(y1 - my)));
    float logy0 = fmaxf(y0 - my - lsy, LOG1EM8);
    float logy1 = fmaxf(y1 - my - lsy, LOG1EM8);

    // ---- Sinkhorn iterations, closed form for C = I:
    // u_new[j] = eps*logx[j] - v[j] - m_u - eps*log(S_u - a_j*(1 - e^{-10}))
    // v_new[i] = eps*logy[i] - u_new[i] - m_v - eps*log(S_v - b_i*(1 - e^{-10}))
    float u0 = 0.f, u1 = 0.f, v0 = 0.f, v1 = 0.f;
    for (int it = 0; it < MAX_ITER; ++it) {
      float mu = wave_max32(fmaxf(u0, u1));
      float a0 = __expf((u0 - mu) * INV_EPS);
      float a1 = __expf((u1 - mu) * INV_EPS);
      float Su = wave_sum32(a0 + a1);
      float un0 = EPS_F * logx0 - v0 - mu - EPS_F * __logf(Su - a0 * ONE_M_EM10);
      float un1 = EPS_F * logx1 - v1 - mu - EPS_F * __logf(Su - a1 * ONE_M_EM10);

      float mv = wave_max32(fmaxf(v0, v1));
      float b0 = __expf((v0 - mv) * INV_EPS);
      float b1 = __expf((v1 - mv) * INV_EPS);
      float Sv = wave_sum32(b0 + b1);
      float vn0 = EPS_F * logy0 - un0 - mv - EPS_F * __logf(Sv - b0 * ONE_M_EM10);
      float vn1 = EPS_F * logy1 - un1 - mv - EPS_F * __logf(Sv - b1 * ONE_M_EM10);

      float d = wave_max32(fmaxf(fabsf(un0 - u0), fabsf(un1 - u1)));
      u0 = un0; u1 = un1; v0 = vn0; v1 = vn1;
      if (d < TOL_F) break;   // wave-uniform after butterfly max
    }

    // ---- wdist = sum_i pi[i,i] / (sum_ij pi + 1e-8),
    // pi = exp(clip((u_i + v_j - delta_ij)/eps, -10, 10)). Clip breaks the
    // rank-1 factorization, so do the explicit 64x64 sum; v[k] broadcasts via
    // v_readlane (uniform k).
    float total = 0.f, diag = 0.f;
    for (int k = 0; k < 32; ++k) {
      float vk0 = bcast_lane(v0, k);   // v[k]
      float vk1 = bcast_lane(v1, k);   // v[k+32]
      float dd  = (lane == k) ? 1.0f : 0.0f;
      float e00 = __expf(fminf(fmaxf((u0 + vk0 - dd) * INV_EPS, -10.f), 10.f));
      float e01 = __expf(fminf(fmaxf((u0 + vk1)      * INV_EPS, -10.f), 10.f));
      float e10 = __expf(fminf(fmaxf((u1 + vk0)      * INV_EPS, -10.f), 10.f));
      float e11 = __expf(fminf(fmaxf((u1 + vk1 - dd) * INV_EPS, -10.f), 10.f));
      total += e00 + e01 + e10 + e11;
      if (lane == k) diag += e00 + e11;
    }
    total = wave_sum32(total);
    diag  = wave_sum32(diag);
    wrow  = fminf(fmaxf(diag / (total + 1e-8f), 0.0f), 100.0f);
  }

  if (lane == 0) s_w[wv] = wrow;
  __syncthreads();
  if (threadIdx.x == 0) {
    float t = 0.f;
#pragma unroll
    for (int i = 0; i < WAVES_PER_BLOCK; ++i) t += s_w[i];
    block_partial[blockIdx.x] = t;
  }
}

// Deterministic final mean: single block tree-reduces the per-block partials.
__global__ __launch_bounds__(256) void reduce_mean_kernel(
    const float* __restrict__ partial, float* __restrict__ out,
    int nblk, int B) {
  __shared__ float s[256];
  float acc = 0.f;
  for (int i = threadIdx.x; i < nblk; i += 256) acc += partial[i];
  s[threadIdx.x] = acc;
  __syncthreads();
  for (int st = 128; st > 0; st >>= 1) {
    if ((int)threadIdx.x < st) s[threadIdx.x] += s[threadIdx.x + st];
    __syncthreads();
  }
  if (threadIdx.x == 0) out[0] = s[0] / (float)B;
}

extern "C" void kernel_launch(void* const* d_in, const int* in_sizes, int n_in,
                              void* d_out, int out_size, void* d_ws, size_t ws_size,
                              hipStream_t stream) {
  const float* x = (const float*)d_in[0];
  const float* y = (const float*)d_in[1];
  const int B = in_sizes[0] / N_DIM;                 // 8192
  const int nblk = (B + WAVES_PER_BLOCK - 1) / WAVES_PER_BLOCK;  // 1024

  float* partial = (float*)d_ws;                     // nblk floats (4 KB)

  sinkhorn_rows_kernel<<<nblk, 256, 0, stream>>>(x, y, partial, B);
  reduce_mean_kernel<<<1, 256, 0, stream>>>(partial, (float*)d_out, nblk, B);
}